// SampleParticles_36653250904489
// MI455X (gfx1250) — compile-verified
//
#include <hip/hip_runtime.h>

// out[b,c,i] = in[b,c, parent_idx[i]]   (B=2, C=32 -> 64 planes; N_IN=1M, N_OUT=2M)
// Pure bandwidth-bound gather: ~776 MB of traffic -> ~33us at 23.3 TB/s.
// Strategy: plane-group tiling for L2 residency of gather targets, async
// global->LDS DMA for index staging (CDNA5 path), NT b128 stores for the
// streaming output.

#define TPB 256   // 8 wave32s per block
#define VEC 4     // outputs (i-direction) per thread, b128 granularity
#define PT  8     // planes handled per block (index reuse + 32MB L2 tile)

typedef float v4f __attribute__((ext_vector_type(4)));
typedef int   v4i __attribute__((ext_vector_type(4)));

// Address-space-qualified pointer element types for the CDNA5 async builtin:
// signature is (global v4i* src, local v4i* dst, imm offset, imm cpol).
typedef __attribute__((address_space(1))) v4i gas_v4i;  // global
typedef __attribute__((address_space(3))) v4i las_v4i;  // LDS

#if __has_builtin(__builtin_amdgcn_global_load_async_to_lds_b128) && \
    __has_builtin(__builtin_amdgcn_s_wait_asynccnt)
#define HAVE_ASYNC_LDS 1
#pragma message("CDNA5: global_load_async_to_lds_b128 builtin path ENABLED")
#else
#define HAVE_ASYNC_LDS 0
#pragma message("CDNA5: async_to_lds builtin NOT available; direct b128 idx loads")
#endif

__global__ __launch_bounds__(TPB) void SampleParticles_gather(
    const float* __restrict__ in, const int* __restrict__ idx,
    float* __restrict__ out, int n_in, int n_out, int planes)
{
#if HAVE_ASYNC_LDS
    __shared__ __align__(16) int sidx[TPB * VEC];
#endif
    const int tid   = threadIdx.x;
    const int i0    = (blockIdx.x * TPB + tid) * VEC;   // base output column
    const int pbase = blockIdx.y * PT;                  // first plane for block
    if (i0 >= n_out) return;

    // ---- Stage this thread's 4 indices (CDNA5 async DMA to LDS) ----
    v4i v;
#if HAVE_ASYNC_LDS
    {
        __builtin_amdgcn_global_load_async_to_lds_b128(
            (gas_v4i*)(idx + i0),            // global source (AS1, typed v4i*)
            (las_v4i*)&sidx[tid * VEC],      // LDS destination (AS3, typed v4i*)
            /*offset=*/0, /*cpol=*/0);
        __builtin_amdgcn_s_wait_asynccnt(0);
        v = *(const v4i*)&sidx[tid * VEC];   // ds_load_b128
    }
#else
    v = *(const v4i*)(idx + i0);             // global_load_b128
#endif

    const float* __restrict__ ip = in  + (size_t)pbase * (size_t)n_in;
    float*       __restrict__ op = out + (size_t)pbase * (size_t)n_out + i0;

    const int pcount = (planes - pbase < PT) ? (planes - pbase) : PT;

    // 8 planes: 4 temporal gather loads (L2-resident tile) + 1 NT b128 store each.
    #pragma unroll
    for (int p = 0; p < PT; ++p) {
        if (p < pcount) {
            v4f r;
            r.x = ip[v.x];
            r.y = ip[v.y];
            r.z = ip[v.z];
            r.w = ip[v.w];
            // Streaming output: non-temporal so the 512MB write stream does
            // not evict the gather tile from L2.
            __builtin_nontemporal_store(r, (v4f*)op);
        }
        ip += n_in;
        op += n_out;
    }
}

extern "C" void kernel_launch(void* const* d_in, const int* in_sizes, int n_in_arrs,
                              void* d_out, int out_size, void* d_ws, size_t ws_size,
                              hipStream_t stream) {
    const float* in  = (const float*)d_in[0];
    const int*   idx = (const int*)d_in[1];
    float*       out = (float*)d_out;

    const int n_out  = in_sizes[1];               // 2,097,152
    const int planes = out_size / n_out;          // B*C = 64
    const int n_in   = in_sizes[0] / planes;      // 1,048,576

    const int ichunks = (n_out + TPB * VEC - 1) / (TPB * VEC);  // 2048
    const int pgroups = (planes + PT - 1) / PT;                 // 8

    dim3 grid(ichunks, pgroups, 1);
    hipLaunchKernelGGL(SampleParticles_gather, grid, dim3(TPB), 0, stream,
                       in, idx, out, n_in, n_out, planes);
}